// ComputeWindowSignature_11106785427739
// MI455X (gfx1250) — compile-verified
//
#include <hip/hip_runtime.h>

typedef __attribute__((ext_vector_type(2))) float v2f;
typedef __attribute__((ext_vector_type(8))) float v8f;

#define BSZ     256
#define LSZ     2048
#define CSZ     6
#define NW      63      // (2048-64)/32 + 1
#define TSTEPS  63      // WIN-1 increments
#define MSTRIDE 68      // padded K stride (words) -> conflict-free banks
#define MROWS   48      // 36 rows padded to 3 WMMA tiles
#define DROWS   16      // 6 cols padded to 16
#define PSTRIDE 8
#define WAVE_LDS (MROWS*MSTRIDE + DROWS*MSTRIDE + 64*PSTRIDE)  // 3264+1088+512 = 4864 words
#define WPB     2       // waves per block

__global__ __launch_bounds__(64)
void ComputeWindowSignature_11106785427739_kernel(const float* __restrict__ path,
                                                  float* __restrict__ out)
{
    __shared__ __align__(16) float lds[WPB * WAVE_LDS];
    const int lane = threadIdx.x & 31;
    const int wv   = threadIdx.x >> 5;
    const int gw   = blockIdx.x * WPB + wv;      // pair id, exact grid -> no bounds check
    const int b    = gw / NW;
    const int w    = gw - b * NW;

    float* M = &lds[wv * WAVE_LDS];              // M^T: [m=0..47][k=0..67]
    float* D = M + MROWS * MSTRIDE;              // D^T: [n=0..15][k=0..67]
    float* P = D + DROWS * MSTRIDE;              // P:   [t][c], stride 8

    // phase 0: zero D and P regions; stage raw 64x6 window into M region (scratch)
    for (int e = lane; e < DROWS*MSTRIDE + 64*PSTRIDE; e += 32) D[e] = 0.f;
    const size_t base = ((size_t)b * LSZ + (size_t)w * 32) * CSZ;
    for (int e = lane; e < 384; e += 32) M[e] = path[base + e];
    __syncthreads();

    // phase 1: increments d_t[c] = x[t+1][c]-x[t][c], transposed into D
    for (int e = lane; e < TSTEPS * CSZ; e += 32) {
        int t = e / CSZ, c = e - t * CSZ;
        D[c * MSTRIDE + t] = M[e + CSZ] - M[e];
    }
    __syncthreads();

    // phase 2: zero M (pad rows/cols must be 0; garbage*0 -> NaN in WMMA)
    for (int e = lane; e < MROWS * MSTRIDE; e += 32) M[e] = 0.f;
    __syncthreads();

    const size_t po = (size_t)gw * 258;

    // phase 3: prefix sums P_t = sum_{u<t} d_u (lanes 0..5); final sum = s1 output
    if (lane < CSZ) {
        float p = 0.f;
        for (int t = 0; t < TSTEPS; ++t) {
            P[t * PSTRIDE + lane] = p;
            p += D[lane * MSTRIDE + t];
        }
        out[po + lane] = p;                       // s1
    }
    __syncthreads();

    // phase 4: sequential level-2 scan; emit M_t rows for the level-3 GEMM
    const int k1 = lane;                 // owned s2 entry 0..31
    const int i1 = k1 / CSZ, j1 = k1 - i1 * CSZ;
    const int k2 = lane + 32;            // second entry for lanes 0..3 (32..35)
    const int i2 = k2 / CSZ, j2 = k2 - i2 * CSZ;
    const bool has2 = (k2 < 36);
    float s2a = 0.f, s2b = 0.f;
    for (int t = 0; t < TSTEPS; ++t) {
        float di = D[i1 * MSTRIDE + t];
        float dj = D[j1 * MSTRIDE + t];
        float pi = P[t * PSTRIDE + i1];
        M[k1 * MSTRIDE + t] = s2a + (0.5f * pi + (1.f/6.f) * di) * dj;
        s2a += (pi + 0.5f * di) * dj;
        if (has2) {
            float di2 = D[i2 * MSTRIDE + t];
            float dj2 = D[j2 * MSTRIDE + t];
            float pi2 = P[t * PSTRIDE + i2];
            M[k2 * MSTRIDE + t] = s2b + (0.5f * pi2 + (1.f/6.f) * di2) * dj2;
            s2b += (pi2 + 0.5f * di2) * dj2;
        }
    }
    out[po + 6 + k1] = 2.f * s2a;                 // 2*s2
    if (has2) out[po + 6 + k2] = 2.f * s2b;
    __syncthreads();

    // phase 5: s3(36x6) = M^T(36x64) x D(64x6) via 3 tiles x 16 K-chunks of
    // v_wmma_f32_16x16x4_f32. EXEC is all-ones here (converged, exact grid).
    const int half = lane >> 4;
    const int l16  = lane & 15;
    #pragma unroll
    for (int tau = 0; tau < 3; ++tau) {
        v8f acc = (v8f)(0.f);
        #pragma unroll
        for (int kc = 0; kc < 16; ++kc) {
            const int kk = 4 * kc + 2 * half;     // A/B: VGPR0=K(kk), VGPR1=K(kk+1)
            v2f a  = *(const v2f*)&M[(16 * tau + l16) * MSTRIDE + kk];
            v2f bf = *(const v2f*)&D[l16 * MSTRIDE + kk];
            acc = __builtin_amdgcn_wmma_f32_16x16x4_f32(
                      false, a, false, bf, (short)0, acc, false, false);
        }
        #pragma unroll
        for (int r = 0; r < 8; ++r) {
            int mg = 16 * tau + 8 * half + r;     // D-matrix row = flat (i,j)
            if (mg < 36 && l16 < CSZ)
                out[po + 42 + mg * CSZ + l16] = 6.f * acc[r];   // 6*s3
        }
    }
}

extern "C" void kernel_launch(void* const* d_in, const int* in_sizes, int n_in,
                              void* d_out, int out_size, void* d_ws, size_t ws_size,
                              hipStream_t stream) {
    const float* path = (const float*)d_in[0];
    float* out = (float*)d_out;
    const int pairs  = BSZ * NW;                  // 16128
    const int blocks = pairs / WPB;               // 8064
    ComputeWindowSignature_11106785427739_kernel<<<blocks, WPB * 32, 0, stream>>>(path, out);
}